// GCN_Embedding_53781580480525
// MI455X (gfx1250) — compile-verified
//
#include <hip/hip_runtime.h>

#define NNODES 500000
#define NEDGES 16000000
#define FDIM 5

typedef __attribute__((ext_vector_type(2))) float v2f;
typedef __attribute__((ext_vector_type(8))) float v8f;

// ---------------------------------------------------------------------------
// WMMA helper: computes y[n][m] = sum_k W[m][k] * act[n][k]  (+ optional bias[m])
// for 16 nodes per wave, using V_WMMA_F32_16X16X4_F32 (exact fp32 math).
//
// Fragment layouts (wave32, from CDNA5 ISA 7.12.2):
//   A (16x4, f32):  lane l: M=l&15; lanes0-15 hold K={0,1} in v0,v1; lanes16-31 K={2,3}
//   B (4x16, f32):  lane l: N=l&15; lanes0-15 hold K={0,1} in v0,v1; lanes16-31 K={2,3}
//   C/D (16x16):    vgpr r, lane l -> (M = r + 8*(l>>4), N = l&15)
// K=5th term handled by a second WMMA with A2/B2 carrying only K=4.
// ---------------------------------------------------------------------------
__device__ __forceinline__ v8f wmma5(const float* __restrict__ W,   // 5x5 row-major
                                     v2f bfrag, float b4,
                                     const float* __restrict__ bias) // may be null
{
    const int lane = threadIdx.x & 31;
    const int m    = lane & 15;
    const int half = lane >> 4;

    v2f a;  a.x = 0.f; a.y = 0.f;
    v2f a2; a2.x = 0.f; a2.y = 0.f;
    if (m < FDIM) {
        a.x  = W[m * FDIM + (half ? 2 : 0)];
        a.y  = W[m * FDIM + (half ? 3 : 1)];
        a2.x = half ? 0.f : W[m * FDIM + 4];
    }
    v2f b2; b2.x = half ? 0.f : b4; b2.y = 0.f;

    v8f c;
#pragma unroll
    for (int r = 0; r < 8; ++r) c[r] = 0.f;
    if (bias != nullptr && !half) {
#pragma unroll
        for (int r = 0; r < FDIM; ++r) c[r] = bias[r];   // C[m][n] = bias[m], m<5
    }

    v8f d = __builtin_amdgcn_wmma_f32_16x16x4_f32(false, a,  false, bfrag, (short)0, c, false, false);
    d     = __builtin_amdgcn_wmma_f32_16x16x4_f32(false, a2, false, b2,    (short)0, d, false, false);
    return d;
}

// D-fragment (features of node n live in lane n, vgprs 0..4) -> next B-fragment
__device__ __forceinline__ void d_to_bfrag(v8f d, v2f* bf, float* b4)
{
    const int lane = threadIdx.x & 31;
    const int n    = lane & 15;
    const int half = lane >> 4;
    float s0 = __shfl(d[0], n, 32);
    float s1 = __shfl(d[1], n, 32);
    float s2 = __shfl(d[2], n, 32);
    float s3 = __shfl(d[3], n, 32);
    float s4 = __shfl(d[4], n, 32);
    bf->x = half ? s2 : s0;
    bf->y = half ? s3 : s1;
    *b4   = half ? 0.f : s4;
}

__device__ __forceinline__ void relu_d(v8f& d)
{
#pragma unroll
    for (int r = 0; r < 8; ++r) d[r] = fmaxf(d[r], 0.f);
}

// ---------------------------------------------------------------------------
// Kernels
// ---------------------------------------------------------------------------
__global__ void zero_kernel(float* __restrict__ p, int n)
{
    int i = blockIdx.x * blockDim.x + threadIdx.x;
    if (i < n) p[i] = 0.f;
}

// x = h @ W1^T   (first GCN lin transform, raw input features)
__global__ void node_xform_first(const float* __restrict__ h,
                                 const float* __restrict__ W,
                                 float* __restrict__ xout, int nNodes)
{
    const int wave = blockIdx.x * (blockDim.x >> 5) + (threadIdx.x >> 5);
    const int base = wave * 16;
    const int lane = threadIdx.x & 31;
    const int n    = lane & 15;
    const int half = lane >> 4;
    const int node = base + n;
    const int cl   = min(node, nNodes - 1);

    v2f b;
    b.x      = h[cl * FDIM + (half ? 2 : 0)];
    b.y      = h[cl * FDIM + (half ? 3 : 1)];
    float b4 = half ? 0.f : h[cl * FDIM + 4];

    v8f d = wmma5(W, b, b4, nullptr);

    if (!half && node < nNodes) {
#pragma unroll
        for (int r = 0; r < FDIM; ++r) xout[node * FDIM + r] = d[r];
    }
}

// act = relu(agg/max(cnt,1) + bprev);  x = act @ Wnext^T
__global__ void node_mid(const float* __restrict__ agg,
                         const float* __restrict__ cnt,
                         const float* __restrict__ bprev,
                         const float* __restrict__ Wnext,
                         float* __restrict__ xout, int nNodes)
{
    const int wave = blockIdx.x * (blockDim.x >> 5) + (threadIdx.x >> 5);
    const int base = wave * 16;
    const int lane = threadIdx.x & 31;
    const int n    = lane & 15;
    const int half = lane >> 4;
    const int node = base + n;
    const int cl   = min(node, nNodes - 1);

    const float inv = 1.0f / fmaxf(cnt[cl], 1.0f);
    const int k0 = half ? 2 : 0, k1 = half ? 3 : 1;
    v2f b;
    b.x      = fmaxf(agg[cl * FDIM + k0] * inv + bprev[k0], 0.f);
    b.y      = fmaxf(agg[cl * FDIM + k1] * inv + bprev[k1], 0.f);
    float b4 = half ? 0.f : fmaxf(agg[cl * FDIM + 4] * inv + bprev[4], 0.f);

    v8f d = wmma5(Wnext, b, b4, nullptr);

    if (!half && node < nNodes) {
#pragma unroll
        for (int r = 0; r < FDIM; ++r) xout[node * FDIM + r] = d[r];
    }
}

// act3 = relu(agg/max(cnt,1) + b3); then fc1(relu) -> fc2(relu) -> fc3 -> out
__global__ void node_final(const float* __restrict__ agg,
                           const float* __restrict__ cnt,
                           const float* __restrict__ b3,
                           const float* __restrict__ fcW1, const float* __restrict__ fcb1,
                           const float* __restrict__ fcW2, const float* __restrict__ fcb2,
                           const float* __restrict__ fcW3, const float* __restrict__ fcb3,
                           float* __restrict__ out, int nNodes)
{
    const int wave = blockIdx.x * (blockDim.x >> 5) + (threadIdx.x >> 5);
    const int base = wave * 16;
    const int lane = threadIdx.x & 31;
    const int n    = lane & 15;
    const int half = lane >> 4;
    const int node = base + n;
    const int cl   = min(node, nNodes - 1);

    const float inv = 1.0f / fmaxf(cnt[cl], 1.0f);
    const int k0 = half ? 2 : 0, k1 = half ? 3 : 1;
    v2f b;
    b.x      = fmaxf(agg[cl * FDIM + k0] * inv + b3[k0], 0.f);
    b.y      = fmaxf(agg[cl * FDIM + k1] * inv + b3[k1], 0.f);
    float b4 = half ? 0.f : fmaxf(agg[cl * FDIM + 4] * inv + b3[4], 0.f);

    v8f d = wmma5(fcW1, b, b4, fcb1);   // fc1 + bias
    relu_d(d);
    d_to_bfrag(d, &b, &b4);

    d = wmma5(fcW2, b, b4, fcb2);       // fc2 + bias
    relu_d(d);
    d_to_bfrag(d, &b, &b4);

    d = wmma5(fcW3, b, b4, fcb3);       // fc3 + bias (no relu)

    if (!half && node < nNodes) {
#pragma unroll
        for (int r = 0; r < FDIM; ++r) out[node * FDIM + r] = d[r];
    }
}

// Per-edge scatter: agg[dst] += x[src]*w ; optionally cnt[dst] += 1.
// Edge streams are non-temporal (960MB of streaming must not evict the 22MB
// of node data that lives in the 192MB L2). Atomics are hardware f32 L2 atomics.
__global__ void edge_scatter(const long long* __restrict__ ei,
                             const float* __restrict__ ew,
                             const float* __restrict__ x,
                             float* __restrict__ agg,
                             float* __restrict__ cnt,   // null => skip counting
                             int nEdges)
{
    int e = blockIdx.x * blockDim.x + threadIdx.x;
    if (e >= nEdges) return;

    long long s = __builtin_nontemporal_load(&ei[e]);
    long long t = __builtin_nontemporal_load(&ei[(size_t)nEdges + (size_t)e]);
    float     w = __builtin_nontemporal_load(&ew[e]);

    const int src = (int)s;
    const int dst = (int)t;
    const float* __restrict__ xs = x + (size_t)src * FDIM;
    float* __restrict__       ad = agg + (size_t)dst * FDIM;

#pragma unroll
    for (int f = 0; f < FDIM; ++f)
        unsafeAtomicAdd(&ad[f], xs[f] * w);

    if (cnt != nullptr)
        unsafeAtomicAdd(&cnt[dst], 1.0f);
}

// ---------------------------------------------------------------------------
// Launch
// ---------------------------------------------------------------------------
extern "C" void kernel_launch(void* const* d_in, const int* in_sizes, int n_in,
                              void* d_out, int out_size, void* d_ws, size_t ws_size,
                              hipStream_t stream)
{
    const float*     h   = (const float*)d_in[0];
    const long long* ei  = (const long long*)d_in[1];
    const float*     ew  = (const float*)d_in[2];
    const float* W1  = (const float*)d_in[3];
    const float* b1  = (const float*)d_in[4];
    const float* W2  = (const float*)d_in[5];
    const float* b2  = (const float*)d_in[6];
    const float* W3  = (const float*)d_in[7];
    const float* b3  = (const float*)d_in[8];
    const float* fcW1 = (const float*)d_in[9];
    const float* fcb1 = (const float*)d_in[10];
    const float* fcW2 = (const float*)d_in[11];
    const float* fcb2 = (const float*)d_in[12];
    const float* fcW3 = (const float*)d_in[13];
    const float* fcb3 = (const float*)d_in[14];
    float* out = (float*)d_out;

    const int N = NNODES;
    const int E = NEDGES;

    float* xbuf = (float*)d_ws;        // N*5 floats
    float* agg  = xbuf + (size_t)N * FDIM; // N*5 floats
    float* cntb = agg  + (size_t)N * FDIM; // N floats

    const int ZB = 256;
    const int nodeBlocks = (N + 127) / 128;          // 8 waves x 16 nodes per block
    const int edgeBlocks = (E + 255) / 256;

    // init accumulators
    zero_kernel<<<(N * FDIM + ZB - 1) / ZB, ZB, 0, stream>>>(agg, N * FDIM);
    zero_kernel<<<(N + ZB - 1) / ZB, ZB, 0, stream>>>(cntb, N);

    // layer 1
    node_xform_first<<<nodeBlocks, 256, 0, stream>>>(h, W1, xbuf, N);
    edge_scatter<<<edgeBlocks, 256, 0, stream>>>(ei, ew, xbuf, agg, cntb, E);
    node_mid<<<nodeBlocks, 256, 0, stream>>>(agg, cntb, b1, W2, xbuf, N);

    // layer 2
    zero_kernel<<<(N * FDIM + ZB - 1) / ZB, ZB, 0, stream>>>(agg, N * FDIM);
    edge_scatter<<<edgeBlocks, 256, 0, stream>>>(ei, ew, xbuf, agg, nullptr, E);
    node_mid<<<nodeBlocks, 256, 0, stream>>>(agg, cntb, b2, W3, xbuf, N);

    // layer 3 + FC head
    zero_kernel<<<(N * FDIM + ZB - 1) / ZB, ZB, 0, stream>>>(agg, N * FDIM);
    edge_scatter<<<edgeBlocks, 256, 0, stream>>>(ei, ew, xbuf, agg, nullptr, E);
    node_final<<<nodeBlocks, 256, 0, stream>>>(agg, cntb, b3,
                                               fcW1, fcb1, fcW2, fcb2, fcW3, fcb3,
                                               out, N);
}